// BipartiteMessagePassingLayer_48077863912208
// MI455X (gfx1250) — compile-verified
//
#include <hip/hip_runtime.h>
#include <hip/hip_bf16.h>

// ---------------------------------------------------------------------------
// Bipartite message passing, restructured for MI455X (gfx1250):
//   CSR build:  deg -> exclusive scan -> bucket-fill (int atomics only)
//   Gather:     S[o,:] = sum_e w_e * attr[a_e,:], ws[o] = sum w_e (no f32 atomics)
//   agg  = (S @ W_a2o^T + b_a2o*ws) / max(ws,1e-6)        [80000 x 256]
//   proj = agg @ W_proj^T + b_proj                          [80000 x 256]
//   out  = relu(obj @ Wu[:,:256]^T + proj @ Wu[:,256:]^T + b_upd)
// GEMMs: V_WMMA_F32_16X16X4_F32; A tiles staged via global_load_async_to_lds.
// ---------------------------------------------------------------------------

typedef __attribute__((ext_vector_type(2))) float v2f;
typedef __attribute__((ext_vector_type(8))) float v8f;

#define BATCH    8
#define NOBJ     10000
#define NATTR    2000
#define IN_DIM   256
#define OUT_DIM  256
#define ATTR_DIM 128
#define NEDGE    800000
#define M_ROWS   (BATCH * NOBJ)   // 80000, divisible by 16

#define USE_ASYNC_STAGE 1

// ---------------------------------------------------------------------------
// CSR build: per-destination degree count (integer atomics, 800k total)
// ---------------------------------------------------------------------------
__global__ void __launch_bounds__(256)
count_kernel(const int* __restrict__ edge_index, int* __restrict__ deg)
{
    int e = blockIdx.x * 256 + threadIdx.x;
    if (e < NEDGE) atomicAdd(&deg[edge_index[e]], 1);
}

// Single-block exclusive scan over 80000 degrees (Hillis-Steele per 1024 chunk)
__global__ void __launch_bounds__(1024)
scan_kernel(const int* __restrict__ deg, int* __restrict__ startA,
            int* __restrict__ cursor)
{
    __shared__ int buf[1024];
    __shared__ int carry;
    if (threadIdx.x == 0) carry = 0;
    __syncthreads();
    for (int base = 0; base < M_ROWS; base += 1024) {
        const int i = base + threadIdx.x;
        const int x = (i < M_ROWS) ? deg[i] : 0;
        buf[threadIdx.x] = x;
        __syncthreads();
        for (int off = 1; off < 1024; off <<= 1) {
            int v = (threadIdx.x >= off) ? buf[threadIdx.x - off] : 0;
            __syncthreads();
            buf[threadIdx.x] += v;
            __syncthreads();
        }
        const int excl = carry + buf[threadIdx.x] - x;
        if (i < M_ROWS) { startA[i] = excl; cursor[i] = excl; }
        __syncthreads();
        if (threadIdx.x == 1023) carry += buf[1023];
        __syncthreads();
    }
}

__global__ void __launch_bounds__(256)
fill_kernel(const int* __restrict__ edge_index, int* __restrict__ cursor,
            int* __restrict__ ebuf)
{
    int e = blockIdx.x * 256 + threadIdx.x;
    if (e < NEDGE) {
        int slot = atomicAdd(&cursor[edge_index[e]], 1);
        ebuf[slot] = e;
    }
}

// ---------------------------------------------------------------------------
// Gather: one wave per object; 128-dim row held in 4 regs/lane; attrs (8MB)
// are L2-resident so the ~410MB of gather reads stay on-chip.
// ---------------------------------------------------------------------------
__global__ void __launch_bounds__(128)
gather_kernel(const int* __restrict__ startA, const int* __restrict__ ebuf,
              const int* __restrict__ edge_index, const float* __restrict__ ew,
              const float* __restrict__ attrs,
              float* __restrict__ S, float* __restrict__ wsum)
{
    const int o = blockIdx.x * 4 + (threadIdx.x >> 5);
    if (o >= M_ROWS) return;
    const int lane = threadIdx.x & 31;
    const int s0 = startA[o];
    const int s1 = (o + 1 < M_ROWS) ? startA[o + 1] : NEDGE;
    float a0 = 0.f, a1 = 0.f, a2 = 0.f, a3 = 0.f, wacc = 0.f;
    for (int i = s0; i < s1; ++i) {
        const int   e = ebuf[i];
        const int   a = edge_index[NEDGE + e];
        const float w = ew[e];
        const float* arow = attrs + (size_t)a * ATTR_DIM;
        a0 += w * arow[lane];
        a1 += w * arow[lane + 32];
        a2 += w * arow[lane + 64];
        a3 += w * arow[lane + 96];
        wacc += w;
    }
    float* srow = S + (size_t)o * ATTR_DIM;
    srow[lane]      = a0;
    srow[lane + 32] = a1;
    srow[lane + 64] = a2;
    srow[lane + 96] = a3;
    if (lane == 0) wsum[o] = wacc;
}

// ---------------------------------------------------------------------------
// fp32 WMMA GEMM:  Out[M,256] = epilogue( A[M,K] @ W[256,K]^T )
//   - 128-thread block (4 waves) per 16-row tile; full N=256 per block so each
//     A element is read from HBM exactly once; wave w owns columns w*64..+63.
//   - A tile staged to LDS with global_load_async_to_lds_b128 (ASYNCcnt path),
//     +4-float row pad: stride (K+4) % 64 banks = 4 -> conflict-free frags.
//   - A split across two sources at KSPLIT fuses the [obj || proj] concat.
// ---------------------------------------------------------------------------
template<int EPILOGUE, int K, int KSPLIT, int LDA0, int LDA1, int LDW>
__global__ void __launch_bounds__(128)
wmma_gemm_kernel(const float* __restrict__ A0,
                 const float* __restrict__ A1,    // used for k >= KSPLIT
                 const float* __restrict__ W,     // [256, LDW] row-major
                 const float* __restrict__ bias,  // [256]
                 const float* __restrict__ wsum,  // [M] (EPILOGUE==0 only)
                 float* __restrict__ Out)         // [M, 256]
{
    constexpr int KP = K + 4;                 // padded LDS row stride (floats)
    constexpr int N  = 256;
    __shared__ float ldsA[16 * KP];

    const int m0   = blockIdx.x * 16;
    const int wid  = threadIdx.x >> 5;        // wave 0..3 -> n0 = wid*64
    const int lane = threadIdx.x & 31;
    const int half = lane >> 4;               // 0 or 1
    const int l15  = lane & 15;
    const int n0   = wid * 64;

    // ---- stage 16 x K A-tile into LDS, 16B per async op, coalesced ----
    constexpr int NV4 = (16 * K) / 4;         // float4 groups
    for (int f = threadIdx.x; f < NV4; f += 128) {
        const int r = (f * 4) / K;
        const int c = (f * 4) % K;
        const float* src;
        if (c < KSPLIT) src = A0 + (size_t)(m0 + r) * LDA0 + c;
        else            src = A1 + (size_t)(m0 + r) * LDA1 + (c - KSPLIT);
#if USE_ASYNC_STAGE
        // ISA 10.2: flat addr low 32 bits == wave-relative LDS byte offset
        unsigned loff = (unsigned)(uintptr_t)&ldsA[r * KP + c];
        unsigned long long ga = (unsigned long long)(uintptr_t)src;
        asm volatile("global_load_async_to_lds_b128 %0, %1, off"
                     :: "v"(loff), "v"(ga) : "memory");
#else
        const float4 v = *(const float4*)src;
        *(float4*)&ldsA[r * KP + c] = v;
#endif
    }
#if USE_ASYNC_STAGE
    asm volatile("s_wait_asynccnt 0x0" ::: "memory");
#endif
    __syncthreads();

    v8f acc[4] = {};   // 4 N-subtiles of 16x16 f32 per wave

    // ---- K loop: 4 columns per V_WMMA_F32_16X16X4_F32 ----
    for (int k0 = 0; k0 < K; k0 += 4) {
        const int kk = k0 + half * 2;
        // A frag: lanes 0-15 -> (row=l15, k=k0..k0+1); lanes 16-31 -> k0+2..+3
        v2f a;
        a.x = ldsA[l15 * KP + kk];
        a.y = ldsA[l15 * KP + kk + 1];
        #pragma unroll
        for (int t = 0; t < 4; ++t) {
            // B frag: B[k][n] = W[n][k]; W is <=512KB -> hot in WGP$/L2
            const int n = n0 + t * 16 + l15;
            v2f b;
            b.x = W[(size_t)n * LDW + kk];
            b.y = W[(size_t)n * LDW + kk + 1];
            acc[t] = __builtin_amdgcn_wmma_f32_16x16x4_f32(
                false, a, false, b, (short)0, acc[t], false, false);
        }
    }

    // ---- epilogue + store ----
    // C/D layout: lane<16: N=..+l15, VGPR v -> M=m0+v ; lanes 16-31 -> M=m0+v+8
    #pragma unroll
    for (int t = 0; t < 4; ++t) {
        const int n = n0 + t * 16 + l15;
        const float bn = bias[n];
        #pragma unroll
        for (int v = 0; v < 8; ++v) {
            const int m = m0 + v + half * 8;
            float val = acc[t][v];
            if (EPILOGUE == 0) {                 // weighted-mean + bias fold
                const float s   = wsum[m];
                const float inv = 1.0f / fmaxf(s, 1e-6f);
                val = (val + bn * s) * inv;
            } else if (EPILOGUE == 1) {          // + bias
                val += bn;
            } else {                             // + bias, ReLU
                val = fmaxf(val + bn, 0.0f);
            }
            Out[(size_t)m * N + n] = val;
        }
    }
}

// ---------------------------------------------------------------------------
// Launch
// ---------------------------------------------------------------------------
extern "C" void kernel_launch(void* const* d_in, const int* in_sizes, int n_in,
                              void* d_out, int out_size, void* d_ws, size_t ws_size,
                              hipStream_t stream) {
    const float* obj        = (const float*)d_in[0];   // [8,10000,256]
    const float* attrs      = (const float*)d_in[1];   // [8,2000,128]
    const int*   edge_index = (const int*)  d_in[2];   // [2,E]
    const float* ew         = (const float*)d_in[3];   // [E]
    const float* W_a2o      = (const float*)d_in[4];   // [256,128]
    const float* b_a2o      = (const float*)d_in[5];   // [256]
    const float* W_proj     = (const float*)d_in[6];   // [256,256]
    const float* b_proj     = (const float*)d_in[7];   // [256]
    const float* W_upd      = (const float*)d_in[8];   // [256,512]
    const float* b_upd      = (const float*)d_in[9];   // [256]
    float* out = (float*)d_out;

    // workspace layout
    int*   deg    = (int*)d_ws;                          // 80000
    int*   startA = deg + M_ROWS;                        // 80000
    int*   cursor = startA + M_ROWS;                     // 80000
    int*   ebuf   = cursor + M_ROWS;                     // 800000
    float* S      = (float*)(ebuf + NEDGE);              // 80000*128 (16B aligned)
    float* wsum   = S + (size_t)M_ROWS * ATTR_DIM;       // 80000
    float* agg    = wsum + M_ROWS;                       // 80000*256
    float* proj   = agg + (size_t)M_ROWS * IN_DIM;       // 80000*256

    hipMemsetAsync(deg, 0, (size_t)M_ROWS * sizeof(int), stream);

    const unsigned egrid = (NEDGE + 255) / 256;
    count_kernel<<<egrid, 256, 0, stream>>>(edge_index, deg);
    scan_kernel<<<1, 1024, 0, stream>>>(deg, startA, cursor);
    fill_kernel<<<egrid, 256, 0, stream>>>(edge_index, cursor, ebuf);
    gather_kernel<<<(M_ROWS + 3) / 4, 128, 0, stream>>>(startA, ebuf, edge_index,
                                                        ew, attrs, S, wsum);

    dim3 gblk(128), ggrid(M_ROWS / 16);
    // agg = (S @ W_a2o^T + b*ws) / max(ws,1e-6)          K=128
    wmma_gemm_kernel<0, ATTR_DIM, ATTR_DIM, ATTR_DIM, ATTR_DIM, ATTR_DIM>
        <<<ggrid, gblk, 0, stream>>>(S, S, W_a2o, b_a2o, wsum, agg);
    // proj = agg @ W_proj^T + b_proj                     K=256
    wmma_gemm_kernel<1, IN_DIM, IN_DIM, IN_DIM, IN_DIM, IN_DIM>
        <<<ggrid, gblk, 0, stream>>>(agg, agg, W_proj, b_proj, nullptr, proj);
    // out = relu([obj || proj] @ W_upd^T + b_upd)        K=512, split at 256
    wmma_gemm_kernel<2, IN_DIM + OUT_DIM, IN_DIM, IN_DIM, OUT_DIM, IN_DIM + OUT_DIM>
        <<<ggrid, gblk, 0, stream>>>(obj, proj, W_upd, b_upd, nullptr, out);
}